// GaussianAgg_66743791780206
// MI455X (gfx1250) — compile-verified
//
#include <hip/hip_runtime.h>
#include <cstdint>

// GaussianAgg for MI455X (gfx1250): bandwidth-bound streaming argmax.
// Noise (264 MB) is staged through LDS with CDNA5 async-to-LDS DMA
// (ASYNCcnt double buffering); compute is thread-per-pixel fp32.
// Sample loop is rolled to keep VGPR pressure low (occupancy), and the
// one-hot mean is accumulated in a per-thread LDS u8 histogram.

#define NB_SAMPLES 16
#define K_CH 32
#define C_CH 33               // K + background
#define TPB 128               // 4 wave32 waves per block
#define SLAB_FLOATS (TPB * C_CH)        // 4224 floats = 16896 bytes per sample-slab
#define CHUNK128 (TPB * 16)             // bytes covered per b128 pass (2048)
#define GAMMA 0.04f
#define EPSF 1e-10f

// ---- CDNA5 async-tensor path: GLOBAL_LOAD_ASYNC_TO_LDS (tracked by ASYNCcnt) ----
__device__ __forceinline__ void async_ld_b128(uint32_t lds_off, const void* gaddr) {
  asm volatile("global_load_async_to_lds_b128 %0, %1, off"
               :: "v"(lds_off), "v"(gaddr) : "memory");
}
__device__ __forceinline__ void async_ld_b32(uint32_t lds_off, const void* gaddr) {
  asm volatile("global_load_async_to_lds_b32 %0, %1, off"
               :: "v"(lds_off), "v"(gaddr) : "memory");
}

// Stage one sample-slab (TPB pixels x 33 ch, 16896 B, 16B aligned) as flat bytes.
// Each wave issues exactly 9 async instructions -> constant s_wait_asynccnt math.
__device__ __forceinline__ void issue_slab(const float* gslab, const void* lslab, int tid) {
  // low 32 bits of a generic pointer derived from LDS == wave-relative LDS byte offset
  uint32_t loff = (uint32_t)(uintptr_t)lslab;
  const char* g = (const char*)gslab;
#pragma unroll
  for (int k = 0; k < 8; ++k)
    async_ld_b128(loff + (uint32_t)(tid * 16 + k * CHUNK128), g + tid * 16 + k * CHUNK128);
  async_ld_b32(loff + (uint32_t)(8 * CHUNK128 + tid * 4), g + 8 * CHUNK128 + tid * 4);
}

__global__ __launch_bounds__(TPB)
void GaussianAgg_66743791780206_kernel(const float* __restrict__ zbuf,
                                       const float* __restrict__ zfar_p,
                                       const float* __restrict__ znear_p,
                                       const float* __restrict__ prob,
                                       const float* __restrict__ mask,
                                       const float* __restrict__ noise,
                                       float* __restrict__ out, int P) {
  __shared__ __align__(16) float lds[2][SLAB_FLOATS];          // 33,792 B double buffer
  __shared__ __align__(16) unsigned char cnt[TPB * C_CH];      // 4,224 B per-thread histogram
  const int tid = threadIdx.x;
  const long long p = (long long)blockIdx.x * TPB + tid;

  // ---- prologue: kick DMA for samples 0 and 1 (overlaps with z_map prep) ----
  const float* nbase = noise + (long long)blockIdx.x * TPB * C_CH;
  const long long sstride = (long long)P * C_CH;
  issue_slab(nbase, &lds[0][0], tid);
  issue_slab(nbase + sstride, &lds[1][0], tid);

  // zero the histogram (dword-wise; ordered before use by the s=0 barrier)
  {
    unsigned int* cz = (unsigned int*)cnt;
    for (int i = tid; i < (TPB * C_CH) / 4; i += TPB) cz[i] = 0u;
  }

  // ---- per-pixel z_map[33] in registers (fp32, matches reference math) ----
  const float zf = zfar_p[0];
  const float zn = znear_p[0];
  const float invR = 1.0f / (zf - zn);

  float zmap[C_CH];
  {
    float zinv[K_CH];
    const float4* zb4 = (const float4*)(zbuf + p * K_CH);  // 128B-aligned rows
    const float4* mk4 = (const float4*)(mask + p * K_CH);
#pragma unroll
    for (int q = 0; q < 8; ++q) {
      float4 z = zb4[q], m = mk4[q];
      zinv[q * 4 + 0] = (zf - z.x) * invR * m.x;
      zinv[q * 4 + 1] = (zf - z.y) * invR * m.y;
      zinv[q * 4 + 2] = (zf - z.z) * invR * m.z;
      zinv[q * 4 + 3] = (zf - z.w) * invR * m.w;
    }
    float zmax = EPSF;  // max(max_k z_inv, EPS)
#pragma unroll
    for (int c = 0; c < K_CH; ++c) zmax = fmaxf(zmax, zinv[c]);

    const float4* pm4 = (const float4*)(prob + p * K_CH);
#pragma unroll
    for (int q = 0; q < 8; ++q) {
      float4 pr = pm4[q];
      zmap[q * 4 + 0] = GAMMA * logf(pr.x + 1e-12f) + zinv[q * 4 + 0] - zmax;
      zmap[q * 4 + 1] = GAMMA * logf(pr.y + 1e-12f) + zinv[q * 4 + 1] - zmax;
      zmap[q * 4 + 2] = GAMMA * logf(pr.z + 1e-12f) + zinv[q * 4 + 2] - zmax;
      zmap[q * 4 + 3] = GAMMA * logf(pr.w + 1e-12f) + zinv[q * 4 + 3] - zmax;
    }
    zmap[K_CH] = EPSF - zmax;  // background channel
  }

  // ---- main loop (rolled: low VGPR pressure): ASYNCcnt double buffering ----
  unsigned char* mycnt = &cnt[tid * C_CH];
#pragma unroll 1
  for (int s = 0; s < NB_SAMPLES; ++s) {
    // slabs complete in order; <=9 outstanding => slab s fully in LDS
    if (s < NB_SAMPLES - 1) {
      asm volatile("s_wait_asynccnt 9" ::: "memory");
    } else {
      asm volatile("s_wait_asynccnt 0" ::: "memory");
    }
    __syncthreads();  // other waves' portions of the slab are done too

    const float* nb = &lds[s & 1][tid * C_CH];  // lane stride 33 dw: bank-conflict-free
    float best = fmaf(GAMMA, nb[0], zmap[0]);
    int bidx = 0;
#pragma unroll
    for (int c = 1; c < C_CH; ++c) {
      float v = fmaf(GAMMA, nb[c], zmap[c]);
      bool gt = v > best;         // strict '>' keeps the first max (jnp.argmax ties)
      best = gt ? v : best;
      bidx = gt ? c : bidx;
    }
    mycnt[bidx] += 1;             // exclusive LDS region: no atomics needed

    __syncthreads();  // everyone finished reading buf (s&1) before DMA overwrites it
    if (s + 2 < NB_SAMPLES)
      issue_slab(nbase + (long long)(s + 2) * sstride, &lds[s & 1][0], tid);
  }

  // ---- mean of one-hots: counts / 16 ----
  float* op = out + p * C_CH;
#pragma unroll
  for (int c = 0; c < C_CH; ++c) op[c] = (float)mycnt[c] * 0.0625f;
}

extern "C" void kernel_launch(void* const* d_in, const int* in_sizes, int n_in,
                              void* d_out, int out_size, void* d_ws, size_t ws_size,
                              hipStream_t stream) {
  const float* zbuf  = (const float*)d_in[0];
  const float* zfar  = (const float*)d_in[1];
  const float* znear = (const float*)d_in[2];
  const float* prob  = (const float*)d_in[3];
  const float* mask  = (const float*)d_in[4];
  const float* noise = (const float*)d_in[5];
  float* out = (float*)d_out;

  const int P = in_sizes[0] / K_CH;  // B*H*W = 131072
  const int blocks = P / TPB;        // 1024
  GaussianAgg_66743791780206_kernel<<<blocks, TPB, 0, stream>>>(
      zbuf, zfar, znear, prob, mask, noise, out, P);
}